// GCN_60129542144783
// MI455X (gfx1250) — compile-verified
//
#include <hip/hip_runtime.h>
#include <math.h>

#define N_NODES 100000
#define N_EDGES 800000
#define IN_DIM  128
#define HID     96
#define OUT_DIM 32
#define BN_EPS  1e-5f

typedef __attribute__((ext_vector_type(2))) float v2f;
typedef __attribute__((ext_vector_type(4))) float v4f;
typedef __attribute__((ext_vector_type(8))) float v8f;

// ---------------------------------------------------------------------------
// Repack W[K x NOUT] into K-pair-interleaved layout so a B fragment is one
// 8-byte load:  Wp[(kp*NOUT + n)*2 + j] = W[(2*kp + j)*NOUT + n]
// ---------------------------------------------------------------------------
__global__ void pack_w_k(const float* __restrict__ W, float* __restrict__ Wp,
                         int K, int NOUT) {
    int g = blockIdx.x * blockDim.x + threadIdx.x;
    int total = (K / 2) * NOUT;
    if (g < total) {
        int kp = g / NOUT, n = g % NOUT;
        Wp[(size_t)g * 2 + 0] = W[(size_t)(2 * kp + 0) * NOUT + n];
        Wp[(size_t)g * 2 + 1] = W[(size_t)(2 * kp + 1) * NOUT + n];
    }
}

// ---------------------------------------------------------------------------
// GEMM + fused prologue/epilogue:
//   Ain  = bnscale ? max(0, A*scale[k]+shift[k]) : A      (applied in LDS stage)
//   H    = Ain @ W                                         (V_WMMA_F32_16X16X4_F32)
//   AGG  = bias[c] + dinv[row]^2 * H                       (self-loop + bias init)
// One block = 128 threads = 4 waves; each wave owns 16 rows; NOUT/16 column
// tiles accumulated simultaneously so the A fragment is reused NT times.
// ---------------------------------------------------------------------------
template <int K, int NOUT>
__global__ __launch_bounds__(128) void gcn_gemm_wmma(
    const float* __restrict__ A, const float* __restrict__ Wp,
    float* __restrict__ Hout, float* __restrict__ AGG,
    const float* __restrict__ dinv, const float* __restrict__ bias,
    int nrows,
    const float* __restrict__ bnscale, const float* __restrict__ bnshift)
{
    constexpr int NT = NOUT / 16;     // column tiles
    constexpr int KP = K + 4;         // padded LDS row stride (bank-conflict free)
    __shared__ float ldsA[64 * KP];

    const int tid  = threadIdx.x;
    const int Rblk = blockIdx.x * 64;
    const bool bn  = (bnscale != nullptr);

    // ---- stage A block [64 x K] into LDS (coalesced float4), fused BN+ReLU ----
    constexpr int NV = 64 * (K / 4);
    for (int i = tid; i < NV; i += 128) {
        const int row = i / (K / 4);
        const int c4  = (i % (K / 4)) * 4;
        const int rg  = Rblk + row;
        v4f v = (v4f){0.f, 0.f, 0.f, 0.f};
        if (rg < nrows)
            v = *(const v4f*)(A + (size_t)rg * K + c4);
        if (bn) {
            v.x = fmaxf(0.f, fmaf(v.x, bnscale[c4 + 0], bnshift[c4 + 0]));
            v.y = fmaxf(0.f, fmaf(v.y, bnscale[c4 + 1], bnshift[c4 + 1]));
            v.z = fmaxf(0.f, fmaf(v.z, bnscale[c4 + 2], bnshift[c4 + 2]));
            v.w = fmaxf(0.f, fmaf(v.w, bnscale[c4 + 3], bnshift[c4 + 3]));
        }
        *(v4f*)&ldsA[row * KP + c4] = v;
    }
    __syncthreads();

    const int w    = tid >> 5;        // wave id 0..3 -> 16-row sub-tile
    const int lane = tid & 31;
    const int ln   = lane & 15;       // M index for A frag / N index for B frag
    const int hi   = lane >> 4;       // 0: K=k..k+1, 1: K=k+2..k+3
    const int kk   = hi * 2;

    v8f acc[NT];
#pragma unroll
    for (int t = 0; t < NT; ++t) acc[t] = (v8f){0, 0, 0, 0, 0, 0, 0, 0};

    const float* lrow = &ldsA[(w * 16 + ln) * KP];
    for (int k = 0; k < K; k += 4) {
        // A fragment 16x4: lanes 0-15 hold K=k..k+1, lanes 16-31 hold K=k+2..k+3
        const v2f a = *(const v2f*)&lrow[k + kk];
        const int kpair = (k >> 1) + hi;  // index into packed W pairs
#pragma unroll
        for (int t = 0; t < NT; ++t) {
            const v2f b = *(const v2f*)&Wp[((size_t)kpair * NOUT + t * 16 + ln) * 2];
            acc[t] = __builtin_amdgcn_wmma_f32_16x16x4_f32(
                false, a, false, b, (short)0, acc[t], false, false);
        }
    }

    // ---- fused epilogue: H = h, AGG = bias + dinv^2 * h ----
    const int rbase = Rblk + w * 16 + hi * 8;
#pragma unroll
    for (int r = 0; r < 8; ++r) {
        const int row = rbase + r;
        if (row < nrows) {
            const float di  = dinv[row];
            const float dsq = di * di;
#pragma unroll
            for (int t = 0; t < NT; ++t) {
                const int col = t * 16 + ln;
                const float h = acc[t][r];
                Hout[(size_t)row * NOUT + col] = h;
                AGG[(size_t)row * NOUT + col] = bias[col] + dsq * h;
            }
        }
    }
}

// ---------------------------------------------------------------------------
// Degree / normalization
// ---------------------------------------------------------------------------
__global__ void deg_init_k(float* deg, int n) {
    int g = blockIdx.x * blockDim.x + threadIdx.x;
    if (g < n) deg[g] = 1.0f;  // self loop
}
__global__ void deg_count_k(const int* __restrict__ dst, float* deg, int e) {
    int g = blockIdx.x * blockDim.x + threadIdx.x;
    if (g < e) atomicAdd(&deg[dst[g]], 1.0f);
}
__global__ void deg_rsqrt_k(float* deg, int n) {
    int g = blockIdx.x * blockDim.x + threadIdx.x;
    if (g < n) deg[g] = rsqrtf(deg[g]);  // deg >= 1 always
}

// ---------------------------------------------------------------------------
// Edge scatter: AGG[dst] += dinv[src]*dinv[dst] * H[src], 4 cols per thread
// ---------------------------------------------------------------------------
__global__ void edge_scatter_k(const int* __restrict__ src, const int* __restrict__ dst,
                               const float* __restrict__ dinv, const float* __restrict__ H,
                               float* __restrict__ AGG, int C) {
    const int C4 = C >> 2;
    int g = blockIdx.x * blockDim.x + threadIdx.x;
    int e = g / C4;
    if (e >= N_EDGES) return;
    int cg = (g - e * C4) * 4;
    int s = src[e], d = dst[e];
    float nrm = dinv[s] * dinv[d];
    const v4f h = *(const v4f*)(H + (size_t)s * C + cg);
    float* op = AGG + (size_t)d * C + cg;
    atomicAdd(op + 0, nrm * h.x);
    atomicAdd(op + 1, nrm * h.y);
    atomicAdd(op + 2, nrm * h.z);
    atomicAdd(op + 3, nrm * h.w);
}

// ---------------------------------------------------------------------------
// BatchNorm statistics + finalize into (scale, shift)
// ---------------------------------------------------------------------------
__global__ void zero_k(float* p, int n) {
    int g = blockIdx.x * blockDim.x + threadIdx.x;
    if (g < n) p[g] = 0.0f;
}
__global__ void bn_stats_k(const float* __restrict__ AGG, float* __restrict__ bnsum,
                           float* __restrict__ bnsq, int n, int C) {
    int c = threadIdx.x;  // blockDim.x == C (96 = 3 waves)
    if (c >= C) return;
    float s = 0.f, q = 0.f;
    for (int r = blockIdx.x; r < n; r += gridDim.x) {
        float v = AGG[(size_t)r * C + c];
        s += v;
        q += v * v;
    }
    atomicAdd(&bnsum[c], s);
    atomicAdd(&bnsq[c], q);
}
__global__ void bn_final_k(const float* __restrict__ bnsum, const float* __restrict__ bnsq,
                           const float* __restrict__ gamma, const float* __restrict__ beta,
                           float* __restrict__ scale, float* __restrict__ shift,
                           int n, int C) {
    int c = threadIdx.x;
    if (c < C) {
        float inv_n = 1.0f / (float)n;
        float mean = bnsum[c] * inv_n;
        float var  = bnsq[c] * inv_n - mean * mean;
        float s    = gamma[c] * rsqrtf(var + BN_EPS);
        scale[c] = s;
        shift[c] = beta[c] - mean * s;
    }
}

// ---------------------------------------------------------------------------
// log_softmax over 32 classes: one wave32 per row, lane == class
// ---------------------------------------------------------------------------
__global__ __launch_bounds__(128) void logsoftmax32_k(const float* __restrict__ AGG,
                                                      float* __restrict__ out, int n) {
    int row  = blockIdx.x * 4 + (threadIdx.x >> 5);
    int lane = threadIdx.x & 31;
    if (row >= n) return;
    float v = AGG[(size_t)row * 32 + lane];
    float mx = v;
#pragma unroll
    for (int o = 16; o > 0; o >>= 1) mx = fmaxf(mx, __shfl_xor(mx, o, 32));
    float e = expf(v - mx);
    float s = e;
#pragma unroll
    for (int o = 16; o > 0; o >>= 1) s += __shfl_xor(s, o, 32);
    out[(size_t)row * 32 + lane] = (v - mx) - logf(s);
}

// ---------------------------------------------------------------------------
// Orchestration
// ---------------------------------------------------------------------------
extern "C" void kernel_launch(void* const* d_in, const int* in_sizes, int n_in,
                              void* d_out, int out_size, void* d_ws, size_t ws_size,
                              hipStream_t stream) {
    const float* x   = (const float*)d_in[0];
    const int*   ei  = (const int*)d_in[1];
    const float* W1  = (const float*)d_in[2];
    const float* b1  = (const float*)d_in[3];
    const float* W2  = (const float*)d_in[4];
    const float* b2  = (const float*)d_in[5];
    const float* W3  = (const float*)d_in[6];
    const float* b3  = (const float*)d_in[7];
    const float* g1  = (const float*)d_in[8];
    const float* be1 = (const float*)d_in[9];
    const float* g2  = (const float*)d_in[10];
    const float* be2 = (const float*)d_in[11];
    float* out = (float*)d_out;

    const int* src = ei;
    const int* dst = ei + N_EDGES;

    // workspace carve
    const size_t NH = (size_t)N_NODES * HID;
    float* H     = (float*)d_ws;
    float* AGG   = H + NH;
    float* dinv  = AGG + NH;
    float* bnsum = dinv + N_NODES;
    float* bnsq  = bnsum + HID;
    float* scal  = bnsq + HID;
    float* shft  = scal + HID;
    float* Wp1   = shft + HID;                    // 128*96
    float* Wp2   = Wp1 + (size_t)IN_DIM * HID;    // 96*96
    float* Wp3   = Wp2 + (size_t)HID * HID;       // 96*32

    const int TB = 256;
    const int gemm_blocks = (N_NODES + 63) / 64;  // 1563
    const int edge96 = (N_EDGES * (HID / 4) + TB - 1) / TB;
    const int edge32 = (N_EDGES * (OUT_DIM / 4) + TB - 1) / TB;

    // --- weight repack (tiny) ---
    pack_w_k<<<((IN_DIM / 2) * HID + TB - 1) / TB, TB, 0, stream>>>(W1, Wp1, IN_DIM, HID);
    pack_w_k<<<((HID / 2) * HID + TB - 1) / TB, TB, 0, stream>>>(W2, Wp2, HID, HID);
    pack_w_k<<<((HID / 2) * OUT_DIM + TB - 1) / TB, TB, 0, stream>>>(W3, Wp3, HID, OUT_DIM);

    // --- degrees / dinv ---
    deg_init_k<<<(N_NODES + TB - 1) / TB, TB, 0, stream>>>(dinv, N_NODES);
    deg_count_k<<<(N_EDGES + TB - 1) / TB, TB, 0, stream>>>(dst, dinv, N_EDGES);
    deg_rsqrt_k<<<(N_NODES + TB - 1) / TB, TB, 0, stream>>>(dinv, N_NODES);

    // ================= Layer 1 =================
    gcn_gemm_wmma<IN_DIM, HID><<<gemm_blocks, 128, 0, stream>>>(
        x, Wp1, H, AGG, dinv, b1, N_NODES, nullptr, nullptr);
    edge_scatter_k<<<edge96, TB, 0, stream>>>(src, dst, dinv, H, AGG, HID);
    zero_k<<<1, 2 * HID, 0, stream>>>(bnsum, 2 * HID);
    bn_stats_k<<<512, HID, 0, stream>>>(AGG, bnsum, bnsq, N_NODES, HID);
    bn_final_k<<<1, HID, 0, stream>>>(bnsum, bnsq, g1, be1, scal, shft, N_NODES, HID);

    // ================= Layer 2 (BN1+ReLU fused into A staging) =================
    gcn_gemm_wmma<HID, HID><<<gemm_blocks, 128, 0, stream>>>(
        AGG, Wp2, H, AGG, dinv, b2, N_NODES, scal, shft);
    edge_scatter_k<<<edge96, TB, 0, stream>>>(src, dst, dinv, H, AGG, HID);
    zero_k<<<1, 2 * HID, 0, stream>>>(bnsum, 2 * HID);
    bn_stats_k<<<512, HID, 0, stream>>>(AGG, bnsum, bnsq, N_NODES, HID);
    bn_final_k<<<1, HID, 0, stream>>>(bnsum, bnsq, g2, be2, scal, shft, N_NODES, HID);

    // ================= Layer 3 (BN2+ReLU fused) =================
    // NOTE: layer-3 GEMM reads AGG (96 cols) and writes AGG3 region; since the
    // A tile for a block is fully staged to LDS before any epilogue store, and
    // NOUT (32) < K (96), writing AGG in-place would alias rows read by OTHER
    // blocks. Use H-offset region for the 32-col buffers instead (disjoint).
    float* H3   = H;             // 100000 x 32 fits in H region
    float* AGG3 = H + (size_t)N_NODES * OUT_DIM;  // also inside H region, disjoint from AGG
    gcn_gemm_wmma<HID, OUT_DIM><<<gemm_blocks, 128, 0, stream>>>(
        AGG, Wp3, H3, AGG3, dinv, b3, N_NODES, scal, shft);
    edge_scatter_k<<<edge32, TB, 0, stream>>>(src, dst, dinv, H3, AGG3, OUT_DIM);

    // --- log_softmax -> d_out ---
    logsoftmax32_k<<<(N_NODES + 3) / 4, 128, 0, stream>>>(AGG3, out, N_NODES);
}